// VarConvND_37168646980365
// MI455X (gfx1250) — compile-verified
//
#include <hip/hip_runtime.h>

typedef __attribute__((ext_vector_type(2))) float v2f;
typedef __attribute__((ext_vector_type(8))) float v8f;

#define B_    8
#define CIN   64
#define HH    64
#define WW    64
#define SS    (HH * WW)        // 4096 spatial positions
#define KD    (CIN * 9)        // 576
#define COUT  64

// Branchless im2col element: k -> (c, di, dj); returns patch+bias for valid
// batch rows, bias for OOB taps of valid rows, 0 for dead rows (m>=8).
// Xb is pre-clamped to a valid batch row, so the load is always in-bounds.
__device__ __forceinline__ float a_elem(const float* __restrict__ Xb,
                                        float bsm, bool row_ok,
                                        int i, int j, int k)
{
    int c  = k / 9;
    int r  = k - c * 9;
    int di = r / 3;
    int dj = r - di * 3;
    int y  = i + di - 1;
    int x  = j + dj - 1;
    bool inb = ((unsigned)y < HH) & ((unsigned)x < WW);
    int idx  = inb ? ((c << 12) + (y << 6) + x) : 0;   // safe index
    float v  = Xb[idx];
    return (row_ok & inb) ? (v + bsm) : bsm;           // dead rows: bsm==0
}

// One wave32 per spatial position s.
// D[16x64] = A[16xK] * B[Kx64] via V_WMMA_F32_16X16X4_F32, K=576 in 144 steps,
// 4 N-tiles of 16. Batch rows 0..7 valid; rows 8..15 are exact zeros.
__global__ __launch_bounds__(256) void varconv_wmma_f32(
    const float* __restrict__ X,     // [B, CIN, H, W]
    const float* __restrict__ Wt,    // [S, KD, COUT]
    const float* __restrict__ bias,  // [H, W] -> [S]
    float* __restrict__ out)         // [B, COUT, H, W]
{
    const int wave = threadIdx.x >> 5;
    const int lane = threadIdx.x & 31;
    const int s    = blockIdx.x * 8 + wave;
    const int i    = s >> 6;          // H index
    const int j    = s & 63;          // W index
    const int half = lane >> 4;       // 0 -> K rows {k0,k0+1}; 1 -> {k0+2,k0+3}
    const int lm   = lane & 15;       // A row (M) / B column-in-tile (N)

    const bool  row_ok = (lm < B_);
    const float bsm    = row_ok ? bias[s] : 0.0f;     // bias for live rows only
    const int   mclamp = row_ok ? lm : (B_ - 1);      // keep loads in-bounds
    const float* Xb    = X + (size_t)mclamp * CIN * SS;
    const float* wbase = Wt + (size_t)s * (KD * COUT);

    v8f acc0 = {}, acc1 = {}, acc2 = {}, acc3 = {};

    #pragma unroll 2
    for (int k0 = 0; k0 < KD; k0 += 4) {
        const int ka = k0 + half * 2;

        // ---- A fragment (branchless, EXEC stays all-ones) ----
        v2f a;
        a.x = a_elem(Xb, bsm, row_ok, i, j, ka);
        a.y = a_elem(Xb, bsm, row_ok, i, j, ka + 1);

        // ---- B fragments: one base pointer + immediate offsets; NT stream ----
        const float* wp = wbase + (size_t)ka * COUT + lm;
        v2f b0, b1, b2, b3;
        b0.x = __builtin_nontemporal_load(wp + 0);
        b0.y = __builtin_nontemporal_load(wp + COUT + 0);
        b1.x = __builtin_nontemporal_load(wp + 16);
        b1.y = __builtin_nontemporal_load(wp + COUT + 16);
        b2.x = __builtin_nontemporal_load(wp + 32);
        b2.y = __builtin_nontemporal_load(wp + COUT + 32);
        b3.x = __builtin_nontemporal_load(wp + 48);
        b3.y = __builtin_nontemporal_load(wp + COUT + 48);

        // keep the HBM weight stream ~8 steps (8KB) ahead of consumption
        __builtin_prefetch(wp + 8 * 4 * COUT, 0, 1);

        acc0 = __builtin_amdgcn_wmma_f32_16x16x4_f32(false, a, false, b0, (short)0, acc0, false, false);
        acc1 = __builtin_amdgcn_wmma_f32_16x16x4_f32(false, a, false, b1, (short)0, acc1, false, false);
        acc2 = __builtin_amdgcn_wmma_f32_16x16x4_f32(false, a, false, b2, (short)0, acc2, false, false);
        acc3 = __builtin_amdgcn_wmma_f32_16x16x4_f32(false, a, false, b3, (short)0, acc3, false, false);
    }

    // ---- store: M rows 0..7 live in acc VGPRs 0..7, lanes 0..15 (N=lane) ----
    if (lane < 16) {
        #pragma unroll
        for (int r = 0; r < 8; ++r) {
            out[((size_t)r * COUT + 0  + lane) * SS + s] = acc0[r];
            out[((size_t)r * COUT + 16 + lane) * SS + s] = acc1[r];
            out[((size_t)r * COUT + 32 + lane) * SS + s] = acc2[r];
            out[((size_t)r * COUT + 48 + lane) * SS + s] = acc3[r];
        }
    }
}

extern "C" void kernel_launch(void* const* d_in, const int* in_sizes, int n_in,
                              void* d_out, int out_size, void* d_ws, size_t ws_size,
                              hipStream_t stream) {
    const float* X    = (const float*)d_in[0];  // [8,64,64,64]
    const float* Wt   = (const float*)d_in[1];  // [4096,576,64]
    const float* bias = (const float*)d_in[2];  // [64,64]
    float* out = (float*)d_out;                 // [8,64,64,64]

    dim3 grid(SS / 8);   // 512 blocks x 8 waves = 4096 positions
    dim3 block(256);
    varconv_wmma_f32<<<grid, block, 0, stream>>>(X, Wt, bias, out);
}